// DynamicMeanVFE_33457795236449
// MI455X (gfx1250) — compile-verified
//
#include <hip/hip_runtime.h>
#include <stdint.h>

// ---------------------------------------------------------------------------
// DynamicMeanVFE for MI455X (gfx1250)
//
// Memory-bound scatter-mean: ~0.2-0.3 GB total traffic -> ~10us at 23.3 TB/s.
// No matrix ops exist in this workload, so the CDNA5 win is data movement:
//   * async global->LDS staging (global_load_async_to_lds_b128 + s_wait_asynccnt)
//     to turn the 20B-stride AoS point stream into coalesced 128B transactions
//   * explicit ds_load_b32 reads of the staged tile (DS pipe, not flat pipe)
//   * open-addressing hash table in d_ws (~96 MB, fits in the 192 MB L2)
//   * native no-return float global atomics for the segment sums
//   * wave32-sized blocks (256 threads = 8 waves)
//
// Round-3 fix: volatile generic-pointer reads of the tile lowered to
// flat_load_b32 (histogram ds=0, vmem+5). Read the tile with explicit
// ds_load_b32 inline asm on the 32-bit LDS byte address instead, with an
// in-asm s_wait_dscnt 0 before the values are consumed.
// ---------------------------------------------------------------------------

namespace {
constexpr int   GRID_X   = 704, GRID_Y = 800, GRID_Z = 26;
constexpr int   SCALE_Z  = GRID_Z;                    // 26
constexpr int   SCALE_YZ = GRID_Y * GRID_Z;           // 20800
constexpr int   SCALE_XYZ = GRID_X * GRID_Y * GRID_Z; // 14,643,200
constexpr int   SENTINEL = 4 * SCALE_XYZ;             // 58,572,800 (< 2^31)
constexpr float MINX = 0.0f, MINY = -40.0f, MINZ = -3.0f;
constexpr float VX = 0.1f, VY = 0.1f, VZ = 0.15f;

constexpr int PTS_PER_BLK    = 256;                   // 8 wave32 waves
constexpr int FLOATS_PER_BLK = PTS_PER_BLK * 5;       // 1280 floats = 5 KB
constexpr int CHUNKS_B128    = (FLOATS_PER_BLK * 4) / 16; // 320 x 16B
}

// ---------------------------------------------------------------------------
// Kernel: bin points and scatter into the hash table.
// Each block stages its 256-point (5 KB) tile into LDS with CDNA5 async
// global->LDS b128 transfers, waits on ASYNCcnt, then reads it back through
// the DS pipe with explicit ds_load_b32.
// ---------------------------------------------------------------------------
__global__ void __launch_bounds__(256)
vfe_scatter(const float* __restrict__ pts, int n,
            int* __restrict__ keys, float* __restrict__ sums,
            float* __restrict__ cnt, unsigned ht_mask) {
  __shared__ float tile[FLOATS_PER_BLK];

  const int  tid     = threadIdx.x;
  const long long blkBase = (long long)blockIdx.x * FLOATS_PER_BLK;
  const long long totF    = (long long)n * 5;
  const bool fullTile = (blkBase + FLOATS_PER_BLK) <= totF;

  // Low 32 bits of the generic address of an LDS object == LDS byte offset
  // (aperture rule), which is exactly what async-to-LDS / ds_load consume.
  const uint32_t lbase = (uint32_t)(uintptr_t)(&tile[0]);

  if (fullTile) {
    // Async copy: 320 x b128 chunks, issued across the 8 waves of the block.
    const uint64_t gbase = (uint64_t)(uintptr_t)(pts + blkBase);
    for (int c = tid; c < CHUNKS_B128; c += PTS_PER_BLK) {
      uint32_t la = lbase + (uint32_t)(c * 16);
      uint64_t ga = gbase + (uint64_t)(c * 16);
      asm volatile("global_load_async_to_lds_b128 %0, %1, off"
                   :: "v"(la), "v"(ga) : "memory");
    }
    // Wave-level wait for our own async transfers; the block barrier below
    // makes every wave's LDS writes visible to every consumer.
    asm volatile("s_wait_asynccnt 0" ::: "memory");
  }
  __syncthreads();

  const long long pidx = (long long)blockIdx.x * PTS_PER_BLK + tid;
  if (pidx >= n) return;

  float p0, p1, p2, p3, p4;
  if (fullTile) {
    // Explicit DS-pipe reads (stride-5 DWORDs over 64 banks: conflict-free).
    const uint32_t laddr = lbase + (uint32_t)(tid * 20);
    asm volatile("ds_load_b32 %0, %5 offset:0\n\t"
                 "ds_load_b32 %1, %5 offset:4\n\t"
                 "ds_load_b32 %2, %5 offset:8\n\t"
                 "ds_load_b32 %3, %5 offset:12\n\t"
                 "ds_load_b32 %4, %5 offset:16\n\t"
                 "s_wait_dscnt 0x0"
                 : "=v"(p0), "=v"(p1), "=v"(p2), "=v"(p3), "=v"(p4)
                 : "v"(laddr)
                 : "memory");
  } else {
    const float* g = pts + pidx * 5;
    p0 = g[0]; p1 = g[1]; p2 = g[2]; p3 = g[3]; p4 = g[4];
  }

  const int b  = (int)p0;
  const int cx = (int)floorf((p1 - MINX) / VX);
  const int cy = (int)floorf((p2 - MINY) / VY);
  const int cz = (int)floorf((p3 - MINZ) / VZ);
  const bool in = (cx >= 0) & (cx < GRID_X) & (cy >= 0) & (cy < GRID_Y) &
                  (cz >= 0) & (cz < GRID_Z);
  const int merge = in ? (b * SCALE_XYZ + cx * SCALE_YZ + cy * SCALE_Z + cz)
                       : SENTINEL;

  // Open-addressing insert (Knuth multiplicative hash, linear probe).
  unsigned h = ((unsigned)merge * 2654435761u) & ht_mask;
  int slot;
  for (;;) {
    int prev = atomicCAS(&keys[h], -1, merge);
    if (prev == -1 || prev == merge) { slot = (int)h; break; }
    h = (h + 1) & ht_mask;
  }

  // Segment sums: native f32 global atomics (no-return -> STOREcnt path).
  float* s = sums + (size_t)slot * 4;
  atomicAdd(s + 0, p1);
  atomicAdd(s + 1, p2);
  atomicAdd(s + 2, p3);
  atomicAdd(s + 3, p4);
  atomicAdd(&cnt[slot], 1.0f);
}

// ---------------------------------------------------------------------------
// Kernel: initialize hash table, counter, and output (sentinel rows).
// Reference fills unused rows with mean=0 and decode(SENTINEL) = {4,0,0,0}.
// ---------------------------------------------------------------------------
__global__ void __launch_bounds__(256)
vfe_init(int* __restrict__ keys, float* __restrict__ cnt,
         float* __restrict__ sums, unsigned* __restrict__ counter,
         float* __restrict__ out_mean, int* __restrict__ out_coords,
         unsigned ht_cap, int max_vox) {
  const unsigned i = blockIdx.x * blockDim.x + threadIdx.x;
  if (i == 0) *counter = 0u;
  if (i < ht_cap) {
    keys[i] = -1;
    cnt[i]  = 0.0f;
    float4* s4 = reinterpret_cast<float4*>(sums);
    s4[i] = make_float4(0.0f, 0.0f, 0.0f, 0.0f);
  }
  if ((int)i < max_vox) {
    float4* m4 = reinterpret_cast<float4*>(out_mean);
    m4[i] = make_float4(0.0f, 0.0f, 0.0f, 0.0f);
    int4* c4 = reinterpret_cast<int4*>(out_coords);
    c4[i] = make_int4(4, 0, 0, 0);   // decode(SENTINEL) -> (b=4, z=0, y=0, x=0)
  }
}

// ---------------------------------------------------------------------------
// Kernel: compact occupied hash slots -> (mean, decoded coords).
// ---------------------------------------------------------------------------
__global__ void __launch_bounds__(256)
vfe_compact(const int* __restrict__ keys, const float* __restrict__ sums,
            const float* __restrict__ cnt, unsigned* __restrict__ counter,
            float* __restrict__ out_mean, int* __restrict__ out_coords,
            unsigned ht_cap, int max_vox) {
  const unsigned s = blockIdx.x * blockDim.x + threadIdx.x;
  if (s >= ht_cap) return;
  const int k = keys[s];
  if (k < 0) return;

  const unsigned idx = atomicAdd(counter, 1u);
  if ((int)idx >= max_vox) return;   // cannot happen (<= N uniques), safety only

  const float c   = cnt[s];
  const float inv = 1.0f / fmaxf(c, 1.0f);
  const float* sv = sums + (size_t)s * 4;
  float4* m4 = reinterpret_cast<float4*>(out_mean);
  m4[idx] = make_float4(sv[0] * inv, sv[1] * inv, sv[2] * inv, sv[3] * inv);

  const int b = k / SCALE_XYZ;
  int r = k - b * SCALE_XYZ;
  const int x = r / SCALE_YZ;  r -= x * SCALE_YZ;
  const int y = r / SCALE_Z;
  const int z = r - y * SCALE_Z;
  int4* c4 = reinterpret_cast<int4*>(out_coords);
  c4[idx] = make_int4(b, z, y, x);
}

// ---------------------------------------------------------------------------
extern "C" void kernel_launch(void* const* d_in, const int* in_sizes, int n_in,
                              void* d_out, int out_size, void* d_ws, size_t ws_size,
                              hipStream_t stream) {
  const float* pts = (const float*)d_in[0];
  const int n = in_sizes[0] / 5;                 // 2,000,000 points
  const int max_vox = out_size / 8;              // two [MAX_VOX,4] outputs

  // Size the hash table from ws_size: cap*(4 key + 4 cnt + 16 sums) + counter.
  unsigned ht_cap = 1u << 22;                    // preferred: 4M slots (~96 MB, L2-resident)
  while (ht_cap > (1u << 18) &&
         ((size_t)ht_cap * 24 + 64) > ws_size) ht_cap >>= 1;
  const unsigned ht_mask = ht_cap - 1;

  char* ws = (char*)d_ws;
  int*      keys    = (int*)ws;
  float*    cnt     = (float*)(ws + (size_t)ht_cap * 4);
  float*    sums    = (float*)(ws + (size_t)ht_cap * 8);
  unsigned* counter = (unsigned*)(ws + (size_t)ht_cap * 24);

  float* out_mean   = (float*)d_out;
  int*   out_coords = (int*)((float*)d_out + (size_t)max_vox * 4);

  // 1) init hash table + sentinel-fill output
  {
    unsigned span = ht_cap > (unsigned)max_vox ? ht_cap : (unsigned)max_vox;
    dim3 g((span + 255) / 256), b(256);
    vfe_init<<<g, b, 0, stream>>>(keys, cnt, sums, counter,
                                  out_mean, out_coords, ht_cap, max_vox);
  }
  // 2) async-staged binning + atomic scatter
  {
    dim3 g((n + PTS_PER_BLK - 1) / PTS_PER_BLK), b(PTS_PER_BLK);
    vfe_scatter<<<g, b, 0, stream>>>(pts, n, keys, sums, cnt, ht_mask);
  }
  // 3) compaction + decode
  {
    dim3 g((ht_cap + 255) / 256), b(256);
    vfe_compact<<<g, b, 0, stream>>>(keys, sums, cnt, counter,
                                     out_mean, out_coords, ht_cap, max_vox);
  }
}